// Top1Gate_22067541967284
// MI455X (gfx1250) — compile-verified
//
#include <hip/hip_runtime.h>
#include <hip/hip_bf16.h>
#include <math.h>

#define S_TOK 8192
#define DIM   2048
#define NEXP  64
#define CAP   128
#define SEC   ((size_t)S_TOK * NEXP * CAP)   // 67108864 elements per output tensor

typedef __attribute__((ext_vector_type(2))) float v2f;
typedef __attribute__((ext_vector_type(8))) float v8f;

// ---------------------------------------------------------------------------
// Kernel 0: zero the tiny per-expert accumulators in workspace
// ---------------------------------------------------------------------------
__global__ void init_stats_kernel(float* __restrict__ me_sum, unsigned* __restrict__ ce_cnt) {
    int t = threadIdx.x;
    if (t < NEXP) { me_sum[t] = 0.0f; ce_cnt[t] = 0u; }
}

// ---------------------------------------------------------------------------
// Kernel 1: logits[s,e] = sum_d x[s,d] * wg[e,d]  via V_WMMA_F32_16X16X4_F32
// One wave (32 threads) per 16-token tile; 4 N-tiles cover all 64 experts.
// A layout (16x4 f32): lanes 0-15 hold M=lane K={k,k+1}; lanes 16-31 M=lane-16 K={k+2,k+3}
// B layout (4x16 f32): lanes 0-15 hold N=lane K={k,k+1}; lanes 16-31 N=lane-16 K={k+2,k+3}
// C layout: VGPR r -> M = r + 8*(lane>=16), N = lane&15
// ---------------------------------------------------------------------------
__global__ __launch_bounds__(32) void gemm_logits_kernel(const float* __restrict__ x,
                                                         const float* __restrict__ wg,
                                                         float* __restrict__ logits) {
    const int m0   = blockIdx.x * 16;
    const int lane = threadIdx.x;
    const int half = lane >> 4;      // 0 or 1
    const int l16  = lane & 15;

    v8f acc0 = {}; v8f acc1 = {}; v8f acc2 = {}; v8f acc3 = {};

    const float* xrow = x + (size_t)(m0 + l16) * DIM;
    const float* b0   = wg + (size_t)(0 * 16 + l16) * DIM;
    const float* b1   = wg + (size_t)(1 * 16 + l16) * DIM;
    const float* b2   = wg + (size_t)(2 * 16 + l16) * DIM;
    const float* b3   = wg + (size_t)(3 * 16 + l16) * DIM;
    const int koff = half * 2;

    for (int k0 = 0; k0 < DIM; k0 += 4) {
        v2f a, w0, w1, w2, w3;
        const float* ap = xrow + k0 + koff;
        a.x  = ap[0];               a.y  = ap[1];
        w0.x = b0[k0 + koff];       w0.y = b0[k0 + koff + 1];
        w1.x = b1[k0 + koff];       w1.y = b1[k0 + koff + 1];
        w2.x = b2[k0 + koff];       w2.y = b2[k0 + koff + 1];
        w3.x = b3[k0 + koff];       w3.y = b3[k0 + koff + 1];

        acc0 = __builtin_amdgcn_wmma_f32_16x16x4_f32(false, a, false, w0, (short)0, acc0, false, false);
        acc1 = __builtin_amdgcn_wmma_f32_16x16x4_f32(false, a, false, w1, (short)0, acc1, false, false);
        acc2 = __builtin_amdgcn_wmma_f32_16x16x4_f32(false, a, false, w2, (short)0, acc2, false, false);
        acc3 = __builtin_amdgcn_wmma_f32_16x16x4_f32(false, a, false, w3, (short)0, acc3, false, false);
    }

    #pragma unroll
    for (int r = 0; r < 8; ++r) {
        const size_t row = (size_t)(m0 + r + half * 8) * NEXP;
        logits[row + 0 * 16 + l16] = acc0[r];
        logits[row + 1 * 16 + l16] = acc1[r];
        logits[row + 2 * 16 + l16] = acc2[r];
        logits[row + 3 * 16 + l16] = acc3[r];
    }
}

// ---------------------------------------------------------------------------
// Kernel 2: per-token softmax + argmax; accumulate me (gate sums) and ce (counts)
// ---------------------------------------------------------------------------
__global__ __launch_bounds__(256) void softmax_argmax_kernel(const float* __restrict__ logits,
                                                             float* __restrict__ gates1_s,
                                                             int* __restrict__ idx1_s,
                                                             float* __restrict__ me_sum,
                                                             unsigned* __restrict__ ce_cnt) {
    __shared__ float    sh_me[NEXP];
    __shared__ unsigned sh_ce[NEXP];
    for (int e = threadIdx.x; e < NEXP; e += blockDim.x) { sh_me[e] = 0.0f; sh_ce[e] = 0u; }
    __syncthreads();

    const int s = blockIdx.x * blockDim.x + threadIdx.x;
    const float4* row = (const float4*)(logits + (size_t)s * NEXP);

    float4 rv[16];
    #pragma unroll
    for (int i = 0; i < 16; ++i) rv[i] = row[i];

    float mx = -3.402823466e38f; int am = 0;
    #pragma unroll
    for (int i = 0; i < 16; ++i) {
        float4 v = rv[i];
        if (v.x > mx) { mx = v.x; am = 4 * i + 0; }
        if (v.y > mx) { mx = v.y; am = 4 * i + 1; }
        if (v.z > mx) { mx = v.z; am = 4 * i + 2; }
        if (v.w > mx) { mx = v.w; am = 4 * i + 3; }
    }

    float sum = 0.0f;
    #pragma unroll
    for (int i = 0; i < 16; ++i) {
        float4 v = rv[i];
        sum += __expf(v.x - mx) + __expf(v.y - mx) + __expf(v.z - mx) + __expf(v.w - mx);
    }
    const float rinv = 1.0f / sum;

    idx1_s[s]   = am;
    gates1_s[s] = rinv;          // gate at argmax: exp(0)/sum (pre-capacity, as in reference)
    atomicAdd(&sh_ce[am], 1u);

    #pragma unroll
    for (int i = 0; i < 16; ++i) {
        float4 v = rv[i];
        atomicAdd(&sh_me[4 * i + 0], __expf(v.x - mx) * rinv);
        atomicAdd(&sh_me[4 * i + 1], __expf(v.y - mx) * rinv);
        atomicAdd(&sh_me[4 * i + 2], __expf(v.z - mx) * rinv);
        atomicAdd(&sh_me[4 * i + 3], __expf(v.w - mx) * rinv);
    }
    __syncthreads();

    if (threadIdx.x < NEXP) {
        atomicAdd(&me_sum[threadIdx.x], sh_me[threadIdx.x]);
        atomicAdd(&ce_cnt[threadIdx.x], sh_ce[threadIdx.x]);
    }
}

// ---------------------------------------------------------------------------
// Kernel 3: per-expert token-order rank (cumsum-1). One wave per expert;
// wave32 ballot + popc prefix preserves exact token ordering.
// ---------------------------------------------------------------------------
__global__ __launch_bounds__(32) void locations_kernel(const int* __restrict__ idx1_s,
                                                       int* __restrict__ loc1_s) {
    const int e    = blockIdx.x;
    const int lane = threadIdx.x;
    int base = 0;
    for (int s0 = 0; s0 < S_TOK; s0 += 32) {
        const int s = s0 + lane;
        const bool pred = (idx1_s[s] == e);
        const unsigned m = (unsigned)__ballot(pred);
        if (pred) loc1_s[s] = base + __popc(m & ((1u << lane) - 1u));
        base += __popc(m);
    }
}

// ---------------------------------------------------------------------------
// Kernel 4: stream-zero the entire output (1 + 2*SEC floats) with b128 stores
// ---------------------------------------------------------------------------
__global__ void zero_out_kernel(float4* __restrict__ p4, float* __restrict__ tail) {
    const size_t n4 = (1ull + 2ull * SEC) / 4ull;   // 33554432 (covers [0, 134217728))
    size_t i = (size_t)blockIdx.x * blockDim.x + threadIdx.x;
    const size_t stride = (size_t)gridDim.x * blockDim.x;
    const float4 z = make_float4(0.f, 0.f, 0.f, 0.f);
    for (; i < n4; i += stride) p4[i] = z;
    if (blockIdx.x == 0 && threadIdx.x == 0) *tail = 0.0f;  // last dispatch element
}

// ---------------------------------------------------------------------------
// Kernel 5: l_aux = E * sum_e( me_sum[e] * cnt[e] ) / S^2   -> out[0]
// ---------------------------------------------------------------------------
__global__ __launch_bounds__(64) void laux_kernel(const float* __restrict__ me_sum,
                                                  const unsigned* __restrict__ ce_cnt,
                                                  float* __restrict__ out0) {
    __shared__ float sh[NEXP];
    const int e = threadIdx.x;
    sh[e] = me_sum[e] * (float)ce_cnt[e];
    __syncthreads();
    if (e == 0) {
        float t = 0.0f;
        #pragma unroll
        for (int i = 0; i < NEXP; ++i) t += sh[i];
        *out0 = t * ((float)NEXP / ((float)S_TOK * (float)S_TOK));
    }
}

// ---------------------------------------------------------------------------
// Kernel 6: sparse scatter — one nonzero per surviving token
// ---------------------------------------------------------------------------
__global__ __launch_bounds__(256) void scatter_kernel(const float* __restrict__ gates1_s,
                                                      const int* __restrict__ idx1_s,
                                                      const int* __restrict__ loc1_s,
                                                      float* __restrict__ out) {
    const int s = blockIdx.x * blockDim.x + threadIdx.x;
    if (s >= S_TOK) return;
    const int l = loc1_s[s];
    if (l >= CAP) return;                      // capacity-dropped token: all-zero row
    const size_t base = 1ull + (size_t)s * NEXP * CAP + (size_t)idx1_s[s] * CAP + (size_t)l;
    out[base]       = gates1_s[s];             // combine1_sec
    out[base + SEC] = 1.0f;                    // dispatch_mask (as float)
}

// ---------------------------------------------------------------------------
extern "C" void kernel_launch(void* const* d_in, const int* in_sizes, int n_in,
                              void* d_out, int out_size, void* d_ws, size_t ws_size,
                              hipStream_t stream) {
    const float* x  = (const float*)d_in[0];   // [S, D] fp32
    const float* wg = (const float*)d_in[1];   // [E, D] fp32
    float* out = (float*)d_out;                // [1 + SEC + SEC] fp32

    // Workspace layout (~2.1 MB)
    float*    logits   = (float*)d_ws;                  // S*E
    float*    gates1_s = logits + (size_t)S_TOK * NEXP; // S
    int*      idx1_s   = (int*)(gates1_s + S_TOK);      // S
    int*      loc1_s   = idx1_s + S_TOK;                // S
    float*    me_sum   = (float*)(loc1_s + S_TOK);      // E
    unsigned* ce_cnt   = (unsigned*)(me_sum + NEXP);    // E

    init_stats_kernel<<<1, 128, 0, stream>>>(me_sum, ce_cnt);
    gemm_logits_kernel<<<S_TOK / 16, 32, 0, stream>>>(x, wg, logits);
    softmax_argmax_kernel<<<S_TOK / 256, 256, 0, stream>>>(logits, gates1_s, idx1_s, me_sum, ce_cnt);
    locations_kernel<<<NEXP, 32, 0, stream>>>(idx1_s, loc1_s);
    zero_out_kernel<<<8192, 256, 0, stream>>>((float4*)out, out + 2ull * SEC);
    laux_kernel<<<1, 64, 0, stream>>>(me_sum, ce_cnt, out);
    scatter_kernel<<<S_TOK / 256, 256, 0, stream>>>(gates1_s, idx1_s, loc1_s, out);
}